// CentralAttentiveModule_84559316124522
// MI455X (gfx1250) — compile-verified
//
#include <hip/hip_runtime.h>
#include <hip/hip_bf16.h>

#define NPTS   500000
#define DCH    64
#define NCLUST 10000
#define BN_EPS 1e-5f
#define LDSTR  68   // padded row stride (floats): 16B-aligned rows, bank = 4n+k (conflict-free)

typedef __attribute__((ext_vector_type(2))) float v2f;
typedef __attribute__((ext_vector_type(8))) float v8f;
typedef __attribute__((ext_vector_type(4))) int   v4i;

#if defined(__gfx1250__) && __has_builtin(__builtin_amdgcn_global_load_async_to_lds_b128)
#define HAVE_ASYNC_LDS 1
#else
#define HAVE_ASYNC_LDS 0
#endif

typedef __attribute__((address_space(1))) v4i* gptr_b128;
typedef __attribute__((address_space(3))) v4i* lptr_b128;

// global->LDS copy of 16 bytes: async DMA path on gfx1250 if available,
// else VGPR-staged fallback.
__device__ __forceinline__ void g2l_b128(const float* gsrc, float* ldst) {
#if HAVE_ASYNC_LDS
    gptr_b128 g = (gptr_b128)(unsigned long long)gsrc;
    lptr_b128 l = (lptr_b128)(unsigned int)(unsigned long long)ldst;
    __builtin_amdgcn_global_load_async_to_lds_b128(g, l, 0, 0);
#else
    float4 t = *(const float4*)gsrc;
    *(float4*)ldst = t;
#endif
}

__device__ __forceinline__ void wait_async_fills() {
#if HAVE_ASYNC_LDS
#if __has_builtin(__builtin_amdgcn_s_wait_asynccnt)
    __builtin_amdgcn_s_wait_asynccnt(0);
#else
    asm volatile("s_wait_asynccnt 0" ::: "memory");
#endif
#endif
}

// ---- monotonic float<->uint encoding so atomicMax(u32) == float max ----
__device__ __forceinline__ unsigned int fenc(float f) {
    unsigned int u = __float_as_uint(f);
    return (u & 0x80000000u) ? ~u : (u | 0x80000000u);
}
__device__ __forceinline__ float fdec(unsigned int u) {
    return __uint_as_float((u & 0x80000000u) ? (u ^ 0x80000000u) : ~u);
}
#define ENC_NEG_INF 0x007FFFFFu  // fenc(-inf)

// ------------------------------------------------------------------ init
__global__ __launch_bounds__(256) void init_kernel(unsigned int* __restrict__ qmax_u,
                                                   unsigned int* __restrict__ mmax_u,
                                                   float* __restrict__ denom,
                                                   float* __restrict__ bn_sum,
                                                   float* __restrict__ bn_sumsq) {
    int i = blockIdx.x * 256 + threadIdx.x;
    if (i < NCLUST * DCH) qmax_u[i] = ENC_NEG_INF;
    if (i < NCLUST) { mmax_u[i] = ENC_NEG_INF; denom[i] = 0.f; }
    if (i < DCH) { bn_sum[i] = 0.f; bn_sumsq[i] = 0.f; }
}

// --------------------------------------------------------- fused WMMA pass
// MODE 0: q projection -> segment max into qmax_u
// MODE 1: k projection -> M[i] = dot(q[cluster[i]], k[i]), segment max mmax_u
// MODE 2: v projection -> h = attn*v -> d_out, accumulate BN sum/sumsq
template <int MODE>
__global__ __launch_bounds__(256) void gemm_pass(
    const float* __restrict__ x, const float* __restrict__ W,
    const float* __restrict__ bias, const int* __restrict__ cluster,
    unsigned int* __restrict__ qmax_u, float* __restrict__ Mbuf,
    unsigned int* __restrict__ mmax_u, const float* __restrict__ ebuf,
    const float* __restrict__ denom, float* __restrict__ hout,
    float* __restrict__ bn_sum, float* __restrict__ bn_sumsq) {
    __shared__ float ldsw[64 * LDSTR];
    __shared__ float ldsx[128 * LDSTR];
    __shared__ float ldsb[64];
    __shared__ float s_sum[64];
    __shared__ float s_sq[64];

    const int tid  = threadIdx.x;
    const int lane = tid & 31;
    const int wave = tid >> 5;
    const int base = blockIdx.x * 128;

    // W (64x64) -> LDS, 1024 x b128 (async DMA on gfx1250)
    #pragma unroll
    for (int it = 0; it < 4; ++it) {
        int idx = tid + it * 256;
        int row = idx >> 4, cq = idx & 15;
        g2l_b128(W + idx * 4, &ldsw[row * LDSTR + cq * 4]);
    }
    // 128-row x tile -> LDS, 2048 x b128 (clamped rows)
    #pragma unroll
    for (int it = 0; it < 8; ++it) {
        int idx = tid + it * 256;
        int row = idx >> 4, cq = idx & 15;
        long grow = base + row;
        if (grow > NPTS - 1) grow = NPTS - 1;
        g2l_b128(x + grow * 64 + cq * 4, &ldsx[row * LDSTR + cq * 4]);
    }
    if (tid < 64) {
        ldsb[tid] = bias[tid];
        if (MODE == 2) { s_sum[tid] = 0.f; s_sq[tid] = 0.f; }
    }
    wait_async_fills();
    __syncthreads();

    const int n    = lane & 15;   // column within 16-wide tile / A row
    const int half = lane >> 4;   // K-pair select for A/B, row-half for C/D
    const float* xw = &ldsx[wave * 16 * LDSTR];

    // accumulators pre-loaded with bias (C[m,j] = b[j])
    v8f acc[4];
    #pragma unroll
    for (int jt = 0; jt < 4; ++jt) {
        float b = ldsb[jt * 16 + n];
        v8f t = {b, b, b, b, b, b, b, b};
        acc[jt] = t;
    }
    // K = 64 in 16 steps of 4; out[m,j] = sum_k x[m,k] * W[j,k]
    #pragma unroll
    for (int ks = 0; ks < 16; ++ks) {
        int kk = ks * 4 + half * 2;
        v2f a = *(const v2f*)&xw[n * LDSTR + kk];  // A: M=lane%16, K=kk,kk+1
        #pragma unroll
        for (int jt = 0; jt < 4; ++jt) {
            v2f b = *(const v2f*)&ldsw[(jt * 16 + n) * LDSTR + kk];  // B[k,j]=W[j,k]
            acc[jt] = __builtin_amdgcn_wmma_f32_16x16x4_f32(
                false, a, false, b, (short)0, acc[jt], false, false);
        }
    }

    // C/D layout: VGPR r -> row m = r + 8*half, col j = jt*16 + n
    const int rowbase = base + wave * 16 + half * 8;

    if (MODE == 0) {
        #pragma unroll
        for (int r = 0; r < 8; ++r) {
            int grow = rowbase + r;
            if (grow < NPTS) {
                int c = cluster[grow];
                #pragma unroll
                for (int jt = 0; jt < 4; ++jt)
                    atomicMax(&qmax_u[c * DCH + jt * 16 + n], fenc(acc[jt][r]));
            }
        }
    } else if (MODE == 1) {
        #pragma unroll
        for (int r = 0; r < 8; ++r) {
            int grow = rowbase + r;
            int gc = grow < NPTS ? grow : NPTS - 1;
            int c = cluster[gc];
            float p = 0.f;
            #pragma unroll
            for (int jt = 0; jt < 4; ++jt)
                p += acc[jt][r] * fdec(qmax_u[c * DCH + jt * 16 + n]);
            // reduce across the 16 lanes of each half (same row)
            p += __shfl_xor(p, 1, 32);
            p += __shfl_xor(p, 2, 32);
            p += __shfl_xor(p, 4, 32);
            p += __shfl_xor(p, 8, 32);
            if (n == 0 && grow < NPTS) {
                Mbuf[grow] = p;
                atomicMax(&mmax_u[c], fenc(p));
            }
        }
    } else {  // MODE 2
        float lsum[4] = {0.f, 0.f, 0.f, 0.f};
        float lsq[4]  = {0.f, 0.f, 0.f, 0.f};
        #pragma unroll
        for (int r = 0; r < 8; ++r) {
            int grow = rowbase + r;
            if (grow < NPTS) {
                int c = cluster[grow];
                float at = ebuf[grow] / denom[c];
                #pragma unroll
                for (int jt = 0; jt < 4; ++jt) {
                    float hv = acc[jt][r] * at;
                    hout[(size_t)grow * DCH + jt * 16 + n] = hv;
                    lsum[jt] += hv;
                    lsq[jt] += hv * hv;
                }
            }
        }
        #pragma unroll
        for (int jt = 0; jt < 4; ++jt) {
            atomicAdd(&s_sum[jt * 16 + n], lsum[jt]);
            atomicAdd(&s_sq[jt * 16 + n], lsq[jt]);
        }
        __syncthreads();
        if (tid < 64) {
            atomicAdd(&bn_sum[tid], s_sum[tid]);
            atomicAdd(&bn_sumsq[tid], s_sq[tid]);
        }
    }
}

// ----------------------------------------------------- segment softmax exp
__global__ __launch_bounds__(256) void softmax_e_kernel(
    const float* __restrict__ Mbuf, const unsigned int* __restrict__ mmax_u,
    const int* __restrict__ cluster, float* __restrict__ ebuf,
    float* __restrict__ denom) {
    int i = blockIdx.x * 256 + threadIdx.x;
    if (i < NPTS) {
        int c = cluster[i];
        float e = __expf(Mbuf[i] - fdec(mmax_u[c]));
        ebuf[i] = e;
        atomicAdd(&denom[c], e);
    }
}

// ------------------------------------------------------------- BN finalize
__global__ void bn_finalize_kernel(const float* __restrict__ bn_sum,
                                   const float* __restrict__ bn_sumsq,
                                   const float* __restrict__ gamma,
                                   const float* __restrict__ beta,
                                   float* __restrict__ scale,
                                   float* __restrict__ shift) {
    int j = threadIdx.x;
    if (j < DCH) {
        const float invN = 1.0f / (float)NPTS;
        float mean = bn_sum[j] * invN;
        float var = bn_sumsq[j] * invN - mean * mean;
        float inv = rsqrtf(var + BN_EPS);
        float sc = gamma[j] * inv;
        scale[j] = sc;
        shift[j] = beta[j] - mean * sc;
    }
}

// ------------------------------------------------------- BN apply + ReLU
__global__ __launch_bounds__(256) void bn_apply_kernel(
    float* __restrict__ out, const float* __restrict__ scale,
    const float* __restrict__ shift) {
    __shared__ float s_scale[64], s_shift[64];
    if (threadIdx.x < 64) {
        s_scale[threadIdx.x] = scale[threadIdx.x];
        s_shift[threadIdx.x] = shift[threadIdx.x];
    }
    __syncthreads();
    long idx = (long)blockIdx.x * 256 + threadIdx.x;  // float4 index
    if (idx < (long)NPTS * 16) {
        float4 h = ((float4*)out)[idx];
        int j0 = ((int)(idx & 15)) * 4;
        h.x = fmaxf(h.x * s_scale[j0 + 0] + s_shift[j0 + 0], 0.f);
        h.y = fmaxf(h.y * s_scale[j0 + 1] + s_shift[j0 + 1], 0.f);
        h.z = fmaxf(h.z * s_scale[j0 + 2] + s_shift[j0 + 2], 0.f);
        h.w = fmaxf(h.w * s_scale[j0 + 3] + s_shift[j0 + 3], 0.f);
        ((float4*)out)[idx] = h;
    }
}

// ------------------------------------------------------------------ launch
extern "C" void kernel_launch(void* const* d_in, const int* in_sizes, int n_in,
                              void* d_out, int out_size, void* d_ws, size_t ws_size,
                              hipStream_t stream) {
    (void)in_sizes; (void)n_in; (void)out_size; (void)ws_size;
    // setup_inputs order: pos, x, cluster, Wv, bv, Wk, bk, Wq, bq, gamma, beta
    const float* x       = (const float*)d_in[1];
    const int*   cluster = (const int*)d_in[2];
    const float* Wv      = (const float*)d_in[3];
    const float* bv      = (const float*)d_in[4];
    const float* Wk      = (const float*)d_in[5];
    const float* bk      = (const float*)d_in[6];
    const float* Wq      = (const float*)d_in[7];
    const float* bq      = (const float*)d_in[8];
    const float* gamma   = (const float*)d_in[9];
    const float* beta    = (const float*)d_in[10];
    float* out = (float*)d_out;

    // workspace carve (~6.6 MB total)
    char* w = (char*)d_ws;
    unsigned int* qmax_u = (unsigned int*)w; w += (size_t)NCLUST * DCH * 4;
    unsigned int* mmax_u = (unsigned int*)w; w += (size_t)NCLUST * 4;
    float* denom   = (float*)w; w += (size_t)NCLUST * 4;
    float* Mbuf    = (float*)w; w += (size_t)NPTS * 4;
    float* ebuf    = (float*)w; w += (size_t)NPTS * 4;
    float* bn_sum  = (float*)w; w += 64 * 4;
    float* bn_sqs  = (float*)w; w += 64 * 4;
    float* bn_scl  = (float*)w; w += 64 * 4;
    float* bn_shf  = (float*)w; w += 64 * 4;

    const int gemm_blocks = (NPTS + 127) / 128;  // 3907

    init_kernel<<<(NCLUST * DCH + 255) / 256, 256, 0, stream>>>(
        qmax_u, mmax_u, denom, bn_sum, bn_sqs);

    gemm_pass<0><<<gemm_blocks, 256, 0, stream>>>(
        x, Wq, bq, cluster, qmax_u, Mbuf, mmax_u, ebuf, denom, out, bn_sum, bn_sqs);

    gemm_pass<1><<<gemm_blocks, 256, 0, stream>>>(
        x, Wk, bk, cluster, qmax_u, Mbuf, mmax_u, ebuf, denom, out, bn_sum, bn_sqs);

    softmax_e_kernel<<<(NPTS + 255) / 256, 256, 0, stream>>>(
        Mbuf, mmax_u, cluster, ebuf, denom);

    gemm_pass<2><<<gemm_blocks, 256, 0, stream>>>(
        x, Wv, bv, cluster, qmax_u, Mbuf, mmax_u, ebuf, denom, out, bn_sum, bn_sqs);

    bn_finalize_kernel<<<1, 64, 0, stream>>>(bn_sum, bn_sqs, gamma, beta, bn_scl, bn_shf);

    bn_apply_kernel<<<((long)NPTS * 16 + 255) / 256, 256, 0, stream>>>(out, bn_scl, bn_shf);
}